// VAE_COUPLE_39496519254566
// MI455X (gfx1250) — compile-verified
//
#include <hip/hip_runtime.h>
#include <hip/hip_bf16.h>
#include <math.h>

// Problem constants (match reference)
#define BB   16      // batch
#define TT   64      // time steps
#define VV   50257   // vocab
#define EE   512     // embed
#define HH   1024    // hidden
#define G3H  3072    // 3*H (r,z,n gate rows)
#define I0   1024    // 2*E (layer-0 input width)

typedef __attribute__((ext_vector_type(2))) float v2f;
typedef __attribute__((ext_vector_type(8))) float v8f;

// ---------------------------------------------------------------------------
// Generic fp32 WMMA GEMM:  C = act( A(MxK) @ W(NxK)^T + bias )
// One wave (32 threads) per block. Each block computes a (16*MT) x 64 tile:
// MT A-fragments x 4 B-fragments -> 4*MT accumulators / WMMAs per K-step.
// Software-pipelined one K-step ahead (peeled epilogue keeps the prefetch
// offset affine so W pointers strength-reduce to +32B increments).
// act: 0 = none, 1 = tanh
// perm: 0 -> C[m*N + n]       (row-major, m = t*B+b)
//       1 -> C[(b*T + t)*N+n] (store logits directly in (B,T,V) order)
// M must be a multiple of 16*MT; N ragged-edge masked at load (clamp) + store.
// K must be a multiple of 4 and >= 8 (here: 512 / 1024).
// ---------------------------------------------------------------------------
template <int MT>
__global__ __launch_bounds__(32)
void wmma_gemm_f32(const float* __restrict__ A, const float* __restrict__ W,
                   const float* __restrict__ bias, float* __restrict__ C,
                   int M, int N, int K, int act, int perm)
{
    const int lane = threadIdx.x;
    const int row  = lane & 15;   // M index within subtile (A), N index (B/C)
    const int hi   = lane >> 4;   // selects K pair (A/B), +8 rows (C)
    const int m0   = blockIdx.y * (16 * MT);
    const int n0   = blockIdx.x << 6;

    const v8f zero8 = {0.f,0.f,0.f,0.f,0.f,0.f,0.f,0.f};
    v8f acc[MT][4];
#pragma unroll
    for (int mt = 0; mt < MT; ++mt)
#pragma unroll
        for (int j = 0; j < 4; ++j) acc[mt][j] = zero8;

    // A fragment pointers: lane holds row (m0+16*mt+row), K slots kk+2*hi+{0,1}
    const float* __restrict__ Ap[MT];
#pragma unroll
    for (int mt = 0; mt < MT; ++mt)
        Ap[mt] = A + (size_t)(m0 + mt * 16 + row) * K + (hi << 1);

    // B fragment pointers: lane holds column n, same K slots. Clamp OOB columns
    // to a valid row (their products are discarded by the masked store).
    const float* __restrict__ Wp[4];
#pragma unroll
    for (int j = 0; j < 4; ++j) {
        int n  = n0 + (j << 4) + row;
        int nc = (n < N) ? n : (N - 1);
        Wp[j] = W + (size_t)nc * K + (hi << 1);
    }

    // Software pipeline: prologue loads K-step 0.
    v2f a[MT], b[4];
#pragma unroll
    for (int mt = 0; mt < MT; ++mt) { a[mt].x = Ap[mt][0]; a[mt].y = Ap[mt][1]; }
#pragma unroll
    for (int j = 0; j < 4; ++j)     { b[j].x  = Wp[j][0];  b[j].y  = Wp[j][1];  }

    // Steady state: prefetch kk+4 while multiplying with kk's fragments.
#pragma unroll 2
    for (int kk = 0; kk < K - 4; kk += 4) {
        const int kn = kk + 4;
        v2f an[MT], bn[4];
#pragma unroll
        for (int mt = 0; mt < MT; ++mt) { an[mt].x = Ap[mt][kn]; an[mt].y = Ap[mt][kn + 1]; }
#pragma unroll
        for (int j = 0; j < 4; ++j)     { bn[j].x  = Wp[j][kn];  bn[j].y  = Wp[j][kn + 1];  }

#pragma unroll
        for (int mt = 0; mt < MT; ++mt)
#pragma unroll
            for (int j = 0; j < 4; ++j)
                acc[mt][j] = __builtin_amdgcn_wmma_f32_16x16x4_f32(
                                 /*neg_a=*/false, a[mt], /*neg_b=*/false, b[j],
                                 /*c_mod=*/(short)0, acc[mt][j],
                                 /*reuse_a=*/false, /*reuse_b=*/false);

#pragma unroll
        for (int mt = 0; mt < MT; ++mt) a[mt] = an[mt];
#pragma unroll
        for (int j = 0; j < 4; ++j)     b[j]  = bn[j];
    }

    // Epilogue: final K-step, no prefetch.
#pragma unroll
    for (int mt = 0; mt < MT; ++mt)
#pragma unroll
        for (int j = 0; j < 4; ++j)
            acc[mt][j] = __builtin_amdgcn_wmma_f32_16x16x4_f32(
                             false, a[mt], false, b[j],
                             (short)0, acc[mt][j], false, false);

    // Store: C VGPR v -> (m = m0 + 16*mt + 8*hi + v, n = n0 + 16*j + row)
#pragma unroll
    for (int j = 0; j < 4; ++j) {
        int n = n0 + (j << 4) + row;
        if (n >= N) continue;
        float bv = bias ? bias[n] : 0.f;
#pragma unroll
        for (int mt = 0; mt < MT; ++mt) {
#pragma unroll
            for (int v = 0; v < 8; ++v) {
                int m = m0 + mt * 16 + (hi << 3) + v;
                float val = acc[mt][j][v] + bv;
                if (act == 1) val = tanhf(val);
                size_t idx;
                if (perm) {
                    int bb = m % BB;          // m = t*B + b
                    int tt = m / BB;
                    idx = (size_t)(bb * TT + tt) * (size_t)N + (size_t)n;
                } else {
                    idx = (size_t)m * (size_t)N + (size_t)n;
                }
                C[idx] = val;
            }
        }
    }
}

// ---------------------------------------------------------------------------
// Elementwise GRU gate combine (one timestep):
//   r = sigma(xr+hr); z = sigma(xz+hz); n = tanh(xn + r*hn)
//   h_new = (1-z)*n + z*h ;  h <- h_new ; y[t] <- h_new
// xp points at the t-th slice (B x 3H, includes b_ih); gh includes b_hh.
// ---------------------------------------------------------------------------
__global__ void gru_gate_kernel(const float* __restrict__ xp,
                                const float* __restrict__ gh,
                                float* __restrict__ h,
                                float* __restrict__ y)
{
    int i = blockIdx.x * blockDim.x + threadIdx.x;
    if (i >= BB * HH) return;
    int b = i / HH, j = i % HH;
    size_t base = (size_t)b * G3H + j;
    float xr = xp[base], xz = xp[base + HH], xn = xp[base + 2 * HH];
    float hr = gh[base], hz = gh[base + HH], hn = gh[base + 2 * HH];
    float r = 1.f / (1.f + expf(-(xr + hr)));
    float z = 1.f / (1.f + expf(-(xz + hz)));
    float n = tanhf(xn + r * hn);
    float hnew = (1.f - z) * n + z * h[i];
    h[i] = hnew;
    if (y) y[i] = hnew;
}

// GRU step with h==0: gh reduces to b_hh; h0 = (1-z)*n.
// g = x @ W_ih^T + b_ih  (B x 3H, from the WMMA GEMM)
__global__ void gru_h0_kernel(const float* __restrict__ g,
                              const float* __restrict__ bhh,
                              float* __restrict__ h)
{
    int i = blockIdx.x * blockDim.x + threadIdx.x;
    if (i >= BB * HH) return;
    int b = i / HH, j = i % HH;
    size_t base = (size_t)b * G3H + j;
    float r = 1.f / (1.f + expf(-(g[base] + bhh[j])));
    float z = 1.f / (1.f + expf(-(g[base + HH] + bhh[HH + j])));
    float n = tanhf(g[base + 2 * HH] + r * bhh[2 * HH + j]);
    h[i] = (1.f - z) * n;
}

// x0[b, 0:2E] = [init_input[b], init_input[b]]
__global__ void build_x0_kernel(const float* __restrict__ init_input,
                                float* __restrict__ x0)
{
    int i = blockIdx.x * blockDim.x + threadIdx.x;
    if (i >= BB * I0) return;
    int b = i / I0, c = i % I0;
    x0[i] = init_input[b * EE + (c % EE)];
}

// X0[t, b, :] = [ emb[helper[b,t]], init_input[b] ]   (T*B x 2E)
__global__ void build_X0_kernel(const int* __restrict__ helper,
                                const float* __restrict__ emb,
                                const float* __restrict__ init_input,
                                float* __restrict__ X0)
{
    int i = blockIdx.x * blockDim.x + threadIdx.x;
    if (i >= TT * BB * I0) return;
    int c  = i % I0;
    int tb = i / I0;
    int b  = tb % BB;
    int t  = tb / BB;
    float v;
    if (c < EE) {
        int tok = helper[b * TT + t];          // helper is (B, T)
        v = emb[(size_t)tok * EE + c];
    } else {
        v = init_input[b * EE + (c - EE)];
    }
    X0[i] = v;
}

// ---------------------------------------------------------------------------
extern "C" void kernel_launch(void* const* d_in, const int* in_sizes, int n_in,
                              void* d_out, int out_size, void* d_ws, size_t ws_size,
                              hipStream_t stream)
{
    (void)in_sizes; (void)n_in; (void)out_size; (void)ws_size;

    const float* init_input = (const float*)d_in[0];
    const int*   helper     = (const int*)  d_in[1];
    const float* emb        = (const float*)d_in[2];
    const float* W_ih0      = (const float*)d_in[3];
    const float* W_hh0      = (const float*)d_in[4];
    const float* b_ih0      = (const float*)d_in[5];
    const float* b_hh0      = (const float*)d_in[6];
    const float* W_ih1      = (const float*)d_in[7];
    const float* W_hh1      = (const float*)d_in[8];
    const float* b_ih1      = (const float*)d_in[9];
    const float* b_hh1      = (const float*)d_in[10];
    const float* Wc_w       = (const float*)d_in[11];
    const float* Wc_b       = (const float*)d_in[12];
    float*       out        = (float*)d_out;

    // Workspace layout (floats), ~27.7 MB total
    float* ws = (float*)d_ws;
    float* x0 = ws;                                  // B*2E
    float* X0 = x0 + (size_t)BB * I0;                // T*B*2E
    float* xp = X0 + (size_t)TT * BB * I0;           // T*B*3H (reused per layer)
    float* Y0 = xp + (size_t)TT * BB * G3H;          // T*B*H
    float* Y1 = Y0 + (size_t)TT * BB * HH;           // T*B*H
    float* F  = Y1 + (size_t)TT * BB * HH;           // T*B*E
    float* gh = F  + (size_t)TT * BB * EE;           // B*3H
    float* h0 = gh + (size_t)BB * G3H;               // B*H (layer0 state)
    float* h1 = h0 + (size_t)BB * HH;                // B*H (layer1 state)

    const dim3 wave(32);
    const int MTB = TT * BB;                         // 1024 rows of decode GEMMs
    const int eltB = 256;

    // ---- Phase A: initial hidden state h0 (one GRU step from zero state) ----
    build_x0_kernel<<<(BB * I0 + eltB - 1) / eltB, eltB, 0, stream>>>(init_input, x0);
    wmma_gemm_f32<1><<<dim3(G3H / 64, 1), wave, 0, stream>>>(x0, W_ih0, b_ih0, gh,
                                                             BB, G3H, I0, 0, 0);
    gru_h0_kernel<<<(BB * HH + eltB - 1) / eltB, eltB, 0, stream>>>(gh, b_hh0, h0);
    wmma_gemm_f32<1><<<dim3(G3H / 64, 1), wave, 0, stream>>>(h0, W_ih1, b_ih1, gh,
                                                             BB, G3H, HH, 0, 0);
    gru_h0_kernel<<<(BB * HH + eltB - 1) / eltB, eltB, 0, stream>>>(gh, b_hh1, h1);

    // ---- Phase B: decoder inputs + layer-0 input gates (parallel over T) ----
    build_X0_kernel<<<(TT * BB * I0 + eltB - 1) / eltB, eltB, 0, stream>>>(
        helper, emb, init_input, X0);
    wmma_gemm_f32<2><<<dim3(G3H / 64, MTB / 32), wave, 0, stream>>>(
        X0, W_ih0, b_ih0, xp, MTB, G3H, I0, 0, 0);

    // ---- Phase C: layer-0 recurrent scan (serial in t) ----
    for (int t = 0; t < TT; ++t) {
        wmma_gemm_f32<1><<<dim3(G3H / 64, 1), wave, 0, stream>>>(
            h0, W_hh0, b_hh0, gh, BB, G3H, HH, 0, 0);
        gru_gate_kernel<<<(BB * HH + eltB - 1) / eltB, eltB, 0, stream>>>(
            xp + (size_t)t * BB * G3H, gh, h0, Y0 + (size_t)t * BB * HH);
    }

    // ---- Phase D: layer-1 input gates + recurrent scan ----
    wmma_gemm_f32<2><<<dim3(G3H / 64, MTB / 32), wave, 0, stream>>>(
        Y0, W_ih1, b_ih1, xp, MTB, G3H, HH, 0, 0);
    for (int t = 0; t < TT; ++t) {
        wmma_gemm_f32<1><<<dim3(G3H / 64, 1), wave, 0, stream>>>(
            h1, W_hh1, b_hh1, gh, BB, G3H, HH, 0, 0);
        gru_gate_kernel<<<(BB * HH + eltB - 1) / eltB, eltB, 0, stream>>>(
            xp + (size_t)t * BB * G3H, gh, h1, Y1 + (size_t)t * BB * HH);
    }

    // ---- Phase E: output projection with fused tanh ----
    wmma_gemm_f32<2><<<dim3(EE / 64, MTB / 32), wave, 0, stream>>>(
        Y1, Wc_w, Wc_b, F, MTB, EE, HH, /*act=*/1, 0);

    // ---- Phase F: vocab logits, stored directly in (B,T,V) order ----
    wmma_gemm_f32<2><<<dim3((VV + 63) / 64, MTB / 32), wave, 0, stream>>>(
        F, emb, nullptr, out, MTB, VV, EE, /*act=*/0, /*perm=*/1);
}